// BinarySplitDecoder_44409961840909
// MI455X (gfx1250) — compile-verified
//
#include <hip/hip_runtime.h>
#include <hip/hip_bf16.h>
#include <stdint.h>

// Binary-split tree decoder.
// DEPTH=8, NUM_ALPHAS=127 per row, output 256 floats per row (col 255 == 0).
// Memory-bound streaming kernel: ~1.53 GB traffic -> ~66us floor @ 23.3 TB/s.
// CDNA5 path used: GLOBAL_LOAD_ASYNC_TO_LDS_B128 + s_wait_asynccnt for
// coalesced staging of per-row alphas into LDS (row stride 127 dwords is
// conflict-free mod 64 banks).

#define TREE_DEPTH   8
#define NUM_ALPHAS   127          // 2^(DEPTH-1) - 1
#define OUT_COLS     256          // 2^DEPTH
#define ROWS_PER_BLK 128          // threads per block == rows per block
#define LDS_FLOATS   (ROWS_PER_BLK * NUM_ALPHAS)   // 16256 floats = 65024 B
#define F4_PER_BLK   (LDS_FLOATS / 4)              // 4064 float4 chunks

__device__ __forceinline__ void async_gl_to_lds_b128(unsigned lds_byte_addr,
                                                     unsigned long long gaddr) {
    // GV mode: vdst = LDS byte address (wave-relative), vaddr = 64-bit global.
    asm volatile("global_load_async_to_lds_b128 %0, %1, off"
                 :: "v"(lds_byte_addr), "v"(gaddr)
                 : "memory");
}

__device__ __forceinline__ void wait_asynccnt0() {
    asm volatile("s_wait_asynccnt 0" ::: "memory");
}

__global__ __launch_bounds__(ROWS_PER_BLK)
void BinarySplitDecoder_44409961840909_kernel(const float* __restrict__ alphas,
                                              float* __restrict__ out,
                                              int batch) {
    __shared__ float sA[LDS_FLOATS];

    const int tid = threadIdx.x;
    const long long row0 = (long long)blockIdx.x * ROWS_PER_BLK;

    // ---- Phase 1: async coalesced copy of this block's alphas into LDS ----
    // Block's alpha region: floats [row0*127, (row0+128)*127).
    // row0*127 = blockIdx*16256 is divisible by 4, so float4 tiling is exact
    // and 16-byte aligned in both global memory and LDS.
    const unsigned long long gbase =
        (unsigned long long)(uintptr_t)alphas + (unsigned long long)row0 * NUM_ALPHAS * 4ull;
    const unsigned lbase = (unsigned)(uintptr_t)(&sA[0]);

    const long long total_f4 = ((long long)batch * NUM_ALPHAS) / 4;  // exact: 1M*127 % 4 == 0
    const long long f4_0     = row0 * NUM_ALPHAS / 4;

    #pragma unroll
    for (int i = 0; i < 32; ++i) {
        const int e = i * ROWS_PER_BLK + tid;        // float4 index within block
        if (e < F4_PER_BLK && (f4_0 + e) < total_f4) {
            async_gl_to_lds_b128(lbase + (unsigned)e * 16u,
                                 gbase + (unsigned long long)e * 16ull);
        }
    }
    wait_asynccnt0();     // this wave's async loads complete (LDS written)
    __syncthreads();      // make all waves' LDS writes visible to everyone

    // ---- Phase 2: per-thread tree build from LDS ----
    const long long row = row0 + tid;
    if (row >= batch) return;

    float* __restrict__ orow = out + row * OUT_COLS;
    const float* __restrict__ a = &sA[tid * NUM_ALPHAS];  // stride 127 dwords: bank-conflict-free

    float lvl[64];        // current level, fully unrolled -> lives in VGPRs
    lvl[0] = 1.0f;
    orow[0] = 1.0f;

    // Levels 1..6 (sizes 2..64): compute children in place (descending i so
    // lvl[2i],lvl[2i+1] never clobber an unread parent), then store the level.
    #pragma unroll
    for (int d = 0; d < 6; ++d) {
        const int n = 1 << d;                 // parents at this level
        #pragma unroll
        for (int i = n - 1; i >= 0; --i) {
            const float al = a[(n - 1) + i];  // alpha column (2^d - 1) + i
            const float p  = lvl[i];
            lvl[2 * i]     = p * al;          // left  = level * a
            lvl[2 * i + 1] = p * (1.0f - al); // right = level * (1 - a)
        }
        #pragma unroll
        for (int j = 0; j < 2 * n; ++j)       // level d+1 block starts at col 2n-1
            orow[(2 * n - 1) + j] = lvl[j];
    }

    // Final level (128 leaves): compute and store pairwise, parents = lvl[0..63].
    #pragma unroll
    for (int i = 0; i < 64; ++i) {
        const float al = a[63 + i];           // alpha columns 63..126
        const float p  = lvl[i];
        orow[127 + 2 * i]     = p * al;
        orow[127 + 2 * i + 1] = p * (1.0f - al);
    }

    orow[255] = 0.0f;     // last column never written by the reference loop
}

extern "C" void kernel_launch(void* const* d_in, const int* in_sizes, int n_in,
                              void* d_out, int out_size, void* d_ws, size_t ws_size,
                              hipStream_t stream) {
    const float* alphas = (const float*)d_in[0];
    float* out = (float*)d_out;
    const int batch = in_sizes[0] / NUM_ALPHAS;   // 1,000,000
    const int blocks = (batch + ROWS_PER_BLK - 1) / ROWS_PER_BLK;
    hipLaunchKernelGGL(BinarySplitDecoder_44409961840909_kernel,
                       dim3(blocks), dim3(ROWS_PER_BLK), 0, stream,
                       alphas, out, batch);
}